// WaveNet_12180527251746
// MI455X (gfx1250) — compile-verified
//
#include <hip/hip_runtime.h>

// ---------------------------------------------------------------------------
// WaveNet-ish stack: 10 dilated 1-ch convs (k=2, dil=2^i) + ReLU, then 1x1
// conv 1->256 channels.  Output [16, 256, 15361] f32 = 252 MB: the kernel is
// HBM-store-bound (252 MB / 23.3 TB/s ~= 11 us).  Strategy: fuse everything,
// keep intermediates in LDS, and express the rank-1 output layer as
// V_WMMA_F32_16X16X4_F32 tiles so the 8 rows/lane of D feed dense b32 stores.
// ---------------------------------------------------------------------------

#define NUM_LAYERS 10
#define NUM_CLASSES 256
#define BB 16
#define TT 16384
#define HALO 1023                 // sum_{i<10} 2^i
#define LF (TT - HALO)            // 15361 final columns
#define TL 256                    // output columns per block
#define NH (TL + HALO)            // 1279 staged inputs per block
#define NTHREADS 256              // 8 wave32

typedef __attribute__((ext_vector_type(2))) float v2f;
typedef __attribute__((ext_vector_type(8))) float v8f;

#if defined(__has_builtin)
#if __has_builtin(__builtin_amdgcn_wmma_f32_16x16x4_f32)
#define HAVE_WMMA_F32X4 1
#endif
#endif

// Output stage for one 16(C)x16(L) tile column sweep.  GUARD=false for the
// 60/61 interior column-blocks where every store is in range (no exec masking
// around the store clause at all).
template <bool GUARD>
__device__ __forceinline__ void emit_tiles(const float* __restrict__ h,
                                           const float* __restrict__ out_w,
                                           const float* __restrict__ out_b,
                                           float* __restrict__ out,
                                           int b, int l_base, int tid) {
  const int wave   = tid >> 5;
  const int lane   = tid & 31;
  const int laneN  = lane & 15;      // column within tile
  const int hiHalf = lane >> 4;      // 0: rows 0..7, 1: rows 8..15

  for (int ct = wave; ct < NUM_CLASSES / 16; ct += NTHREADS / 32) {
    const int c0 = ct * 16;

#if HAVE_WMMA_F32X4
    // A (16x4 f32): out_w in the K=0 column, all other K slots zero.  Only
    // this operand needs predication -- with A's K>=1 columns zero, B's K>=1
    // rows are multiplied by 0 and can hold arbitrary finite values.
    v2f a;
    a.x = (hiHalf == 0) ? out_w[c0 + laneN] : 0.0f;
    a.y = 0.0f;
#endif
    float cbias[8];
#pragma unroll
    for (int r = 0; r < 8; ++r)
      cbias[r] = out_b[c0 + r + hiHalf * 8];

    for (int lt = 0; lt < TL / 16; ++lt) {
      const int l0 = lt * 16;
      v8f dmat;
#if HAVE_WMMA_F32X4
      // B (4x16 f32): h row in K=0.  Unpredicated full-wave LDS load: lanes
      // 16..31 drop the same (finite) h values into a K>=1 row slot whose A
      // column is zero, so they contribute exactly 0 to D.
      v2f bv;
      bv.x = h[l0 + laneN];
      bv.y = 0.0f;
      v8f cacc;
#pragma unroll
      for (int r = 0; r < 8; ++r) cacc[r] = cbias[r];
      // D = A x B + C  (rank-1: out_w (x) h, bias in C)
      dmat = __builtin_amdgcn_wmma_f32_16x16x4_f32(
          /*neg_a=*/false, a, /*neg_b=*/false, bv,
          /*c_mod=*/(short)0, cacc, /*reuse_a=*/false, /*reuse_b=*/false);
#else
      const float hv = h[l0 + laneN];
#pragma unroll
      for (int r = 0; r < 8; ++r) {
        const int m = c0 + r + hiHalf * 8;
        dmat[r] = fmaf(out_w[m], hv, cbias[r]);
      }
#endif
      // Store: VGPR r holds row M=r(+8 for hi half), lanes spread N=0..15.
      // Each store covers two contiguous 64B row segments per wave.
      const int gl = l_base + l0 + laneN;
      if (!GUARD || gl < LF) {
#pragma unroll
        for (int r = 0; r < 8; ++r) {
          const int m = c0 + r + hiHalf * 8;
          out[((size_t)(b * NUM_CLASSES + m)) * LF + gl] = dmat[r];
        }
      }
    }
  }
}

__global__ __launch_bounds__(NTHREADS)
void wavenet_fused(const float* __restrict__ x,
                   const float* __restrict__ conv_w,   // [10,2]
                   const float* __restrict__ conv_b,   // [10]
                   const float* __restrict__ out_w,    // [256]
                   const float* __restrict__ out_b,    // [256]
                   float* __restrict__ out) {          // [16,256,15361]
  __shared__ float s0[NH + 1];
  __shared__ float s1[NH + 1];

  const int b      = blockIdx.y;
  const int l_base = blockIdx.x * TL;
  const int tid    = threadIdx.x;

  // ---- Stage x tile + halo into LDS (zero-pad past end of row). ----
  const float* xrow = x + (size_t)b * TT;
  for (int j = tid; j < NH; j += NTHREADS) {
    const int g = l_base + j;
    s0[j] = (g < TT) ? xrow[g] : 0.0f;
  }
  __syncthreads();

  // ---- 10 dilated causal conv layers, ping-pong s0<->s1 in LDS. ----
  // Fully unrolled: dilation and src/dst arrays are compile-time constants.
  int len = NH;
#pragma unroll
  for (int i = 0; i < NUM_LAYERS; ++i) {
    const int   d  = 1 << i;
    const float w0 = conv_w[2 * i + 0];
    const float w1 = conv_w[2 * i + 1];
    const float cb = conv_b[i];
    const float* __restrict__ src = (i & 1) ? s1 : s0;
    float* __restrict__       dst = (i & 1) ? s0 : s1;
    const int olen = len - d;
    for (int p = tid; p < olen; p += NTHREADS) {
      float v = fmaf(w1, src[p + d], fmaf(w0, src[p], cb));
      dst[p] = fmaxf(v, 0.0f);
    }
    len = olen;
    __syncthreads();
  }
  // NUM_LAYERS is even -> final h tile (TL values) lives in s0.
  const float* __restrict__ h = s0;

  // ---- Output layer: out[c,l] = out_w[c]*h[l] + out_b[c] via WMMA tiles. ----
  if (l_base + TL <= LF) {
    emit_tiles<false>(h, out_w, out_b, out, b, l_base, tid);  // interior: no guard
  } else {
    emit_tiles<true>(h, out_w, out_b, out, b, l_base, tid);   // last column block
  }
}

extern "C" void kernel_launch(void* const* d_in, const int* in_sizes, int n_in,
                              void* d_out, int out_size, void* d_ws, size_t ws_size,
                              hipStream_t stream) {
  (void)in_sizes; (void)n_in; (void)out_size; (void)d_ws; (void)ws_size;
  const float* x      = (const float*)d_in[0];
  const float* conv_w = (const float*)d_in[1];
  const float* conv_b = (const float*)d_in[2];
  const float* out_w  = (const float*)d_in[3];
  const float* out_b  = (const float*)d_in[4];
  float* out = (float*)d_out;

  dim3 grid((LF + TL - 1) / TL, BB);  // 61 x 16 = 976 blocks, 8 waves each
  wavenet_fused<<<grid, NTHREADS, 0, stream>>>(x, conv_w, conv_b, out_w, out_b, out);
}